// MessagePassingGNN_34995393527966
// MI455X (gfx1250) — compile-verified
//
#include <hip/hip_runtime.h>
#include <math.h>

typedef __attribute__((ext_vector_type(2))) float v2f;
typedef __attribute__((ext_vector_type(8))) float v8f;

#define D 512
#define NNODE 256
#define NB 2
#define M_ROWS (NB * NNODE)   // 512

__device__ __forceinline__ float sigmoidf_(float x) {
    return 1.0f / (1.0f + expf(-x));
}

// ---------------------------------------------------------------------------
// 1) agg[b,d] = mean_n nf[b,n,d]     grid: 4 x 256 threads (B*D = 1024 outs)
// ---------------------------------------------------------------------------
__global__ void agg_kernel(const float* __restrict__ nf, float* __restrict__ agg) {
    int idx = blockIdx.x * 256 + threadIdx.x;      // 0..1023
    int b = idx >> 9;
    int d = idx & (D - 1);
    const float* p = nf + (size_t)b * NNODE * D + d;
    float s = 0.0f;
    #pragma unroll 4
    for (int n = 0; n < NNODE; ++n) s += p[(size_t)n * D];
    agg[idx] = s * (1.0f / (float)NNODE);
}

// ---------------------------------------------------------------------------
// 2) upd = relu(agg@nu_w1 + nu_b1) @ nu_w2 + nu_b2    grid: 2 blocks x 256
// ---------------------------------------------------------------------------
__global__ void upd_mlp_kernel(const float* __restrict__ agg,
                               const float* __restrict__ w1, const float* __restrict__ b1,
                               const float* __restrict__ w2, const float* __restrict__ b2,
                               float* __restrict__ upd) {
    int b = blockIdx.x;
    __shared__ float s_in[D];
    __shared__ float s_h[D];
    for (int d = threadIdx.x; d < D; d += 256) s_in[d] = agg[b * D + d];
    __syncthreads();
    for (int o = threadIdx.x; o < D; o += 256) {
        float acc = b1[o];
        for (int k = 0; k < D; ++k) acc = fmaf(s_in[k], w1[k * D + o], acc);
        s_h[o] = fmaxf(acc, 0.0f);
    }
    __syncthreads();
    for (int o = threadIdx.x; o < D; o += 256) {
        float acc = b2[o];
        for (int k = 0; k < D; ++k) acc = fmaf(s_h[k], w2[k * D + o], acc);
        upd[b * D + o] = acc;
    }
}

// ---------------------------------------------------------------------------
// 3) u = nf + broadcast(upd)        grid: 1024 x 256  (512*512 elements)
// ---------------------------------------------------------------------------
__global__ void build_u_kernel(const float* __restrict__ nf, const float* __restrict__ upd,
                               float* __restrict__ u) {
    int idx = blockIdx.x * 256 + threadIdx.x;      // 0..262143
    int b = idx >> 17;                             // 131072 elems per batch
    u[idx] = nf[idx] + upd[b * D + (idx & (D - 1))];
}

// ---------------------------------------------------------------------------
// 4) C[512,512] = A[512,512] @ B[512,512]  (+bias, +relu optional)
//    fp32 WMMA 16x16x4. Block = 256 threads = 8 waves; wave -> 16(M) x 64(N)
//    strip (4 accumulators); block tile 128x64; grid (4, 8).
//    A frag (16x4):  lanes 0-15 row=l, k=k0..k0+1 ; lanes 16-31 row=l, k=k0+2..k0+3
//    B frag (4x16):  lanes 0-15 col=l, k=k0,k0+1  ; lanes 16-31 col=l, k=k0+2,k0+3
//    C/D (16x16):    VGPR r -> row r (lanes 0-15) / row r+8 (lanes 16-31), col=l
// ---------------------------------------------------------------------------
__global__ void gemm512_wmma_kernel(const float* __restrict__ A, const float* __restrict__ Bm,
                                    const float* __restrict__ bias, float* __restrict__ C,
                                    int do_relu) {
    const int lane = threadIdx.x & 31;
    const int wave = threadIdx.x >> 5;
    const int half = lane >> 4;          // 0: lanes 0-15, 1: lanes 16-31
    const int l16  = lane & 15;
    const int m0 = blockIdx.x * 128 + wave * 16;
    const int n0 = blockIdx.y * 64;

    v8f acc0 = {}, acc1 = {}, acc2 = {}, acc3 = {};
    const float* arow = A + (size_t)(m0 + l16) * D;

    for (int k0 = 0; k0 < D; k0 += 4) {
        // A fragment: two consecutive fp32, 8B aligned
        v2f a = *(const v2f*)(arow + k0 + half * 2);
        const float* bbase = Bm + (size_t)(k0 + half * 2) * D + n0 + l16;
        {
            v2f bf; bf.x = bbase[0];      bf.y = bbase[D];
            acc0 = __builtin_amdgcn_wmma_f32_16x16x4_f32(false, a, false, bf, (short)0, acc0, false, false);
        }
        {
            v2f bf; bf.x = bbase[16];     bf.y = bbase[16 + D];
            acc1 = __builtin_amdgcn_wmma_f32_16x16x4_f32(false, a, false, bf, (short)0, acc1, false, false);
        }
        {
            v2f bf; bf.x = bbase[32];     bf.y = bbase[32 + D];
            acc2 = __builtin_amdgcn_wmma_f32_16x16x4_f32(false, a, false, bf, (short)0, acc2, false, false);
        }
        {
            v2f bf; bf.x = bbase[48];     bf.y = bbase[48 + D];
            acc3 = __builtin_amdgcn_wmma_f32_16x16x4_f32(false, a, false, bf, (short)0, acc3, false, false);
        }
    }

    v8f accs[4] = {acc0, acc1, acc2, acc3};
    #pragma unroll
    for (int t = 0; t < 4; ++t) {
        int col = n0 + t * 16 + l16;
        float bv = bias ? bias[col] : 0.0f;
        #pragma unroll
        for (int r = 0; r < 8; ++r) {
            int row = m0 + r + half * 8;
            float v = accs[t][r] + bv;
            if (do_relu) v = fmaxf(v, 0.0f);
            C[(size_t)row * D + col] = v;
        }
    }
}

// ---------------------------------------------------------------------------
// 5) node head: out[m*3+c] = sigmoid(t1[m,:] . nt_w2[:,c] + nt_b2[c])
//    grid: 6 x 256 (1536 outputs)
// ---------------------------------------------------------------------------
__global__ void node_head_kernel(const float* __restrict__ t1, const float* __restrict__ w2,
                                 const float* __restrict__ b2, float* __restrict__ out) {
    int idx = blockIdx.x * 256 + threadIdx.x;
    if (idx >= M_ROWS * 3) return;
    int m = idx / 3, c = idx - m * 3;
    const float* row = t1 + (size_t)m * D;
    float acc = b2[c];
    #pragma unroll 4
    for (int d = 0; d < D; ++d) acc = fmaf(row[d], w2[d * 3 + c], acc);
    out[idx] = sigmoidf_(acc);
}

// ---------------------------------------------------------------------------
// 6) edge head: one block per (b,i); thread j streams hj row.
//    edge[b, i*256+j] = sigmoid( sum_d relu(hi[b,i,d]+hj[b,j,d]+et_b1[d]) * et_w2[d] + et_b2 )
//    grid: 512 x 256
// ---------------------------------------------------------------------------
__global__ void edge_kernel(const float* __restrict__ hi, const float* __restrict__ hj,
                            const float* __restrict__ et_b1, const float* __restrict__ et_w2,
                            const float* __restrict__ et_b2, float* __restrict__ out_edge) {
    int bi = blockIdx.x;            // b*256 + i
    int b  = bi >> 8;
    int j  = threadIdx.x;

    __shared__ float s_hi[D];       // hi row with bias pre-added
    __shared__ float s_w2[D];
    for (int d = threadIdx.x; d < D; d += 256) {
        s_hi[d] = hi[(size_t)bi * D + d] + et_b1[d];
        s_w2[d] = et_w2[d];
    }
    __syncthreads();

    const float* hjrow = hj + (size_t)(b * NNODE + j) * D;
    float acc = et_b2[0];
    #pragma unroll 4
    for (int d = 0; d < D; d += 4) {
        float4 h = *(const float4*)(hjrow + d);
        acc = fmaf(fmaxf(h.x + s_hi[d + 0], 0.0f), s_w2[d + 0], acc);
        acc = fmaf(fmaxf(h.y + s_hi[d + 1], 0.0f), s_w2[d + 1], acc);
        acc = fmaf(fmaxf(h.z + s_hi[d + 2], 0.0f), s_w2[d + 2], acc);
        acc = fmaf(fmaxf(h.w + s_hi[d + 3], 0.0f), s_w2[d + 3], acc);
    }
    out_edge[(size_t)bi * NNODE + j] = sigmoidf_(acc);
}

// ---------------------------------------------------------------------------
// launcher
// ---------------------------------------------------------------------------
extern "C" void kernel_launch(void* const* d_in, const int* in_sizes, int n_in,
                              void* d_out, int out_size, void* d_ws, size_t ws_size,
                              hipStream_t stream) {
    const float* nf    = (const float*)d_in[0];
    const float* nu_w1 = (const float*)d_in[1];
    const float* nu_b1 = (const float*)d_in[2];
    const float* nu_w2 = (const float*)d_in[3];
    const float* nu_b2 = (const float*)d_in[4];
    const float* nt_w1 = (const float*)d_in[5];
    const float* nt_b1 = (const float*)d_in[6];
    const float* nt_w2 = (const float*)d_in[7];
    const float* nt_b2 = (const float*)d_in[8];
    const float* et_w1 = (const float*)d_in[9];
    const float* et_b1 = (const float*)d_in[10];
    const float* et_w2 = (const float*)d_in[11];
    const float* et_b2 = (const float*)d_in[12];

    float* out = (float*)d_out;
    float* out_node = out;                 // 1536 floats
    float* out_edge = out + M_ROWS * 3;    // 131072 floats

    const size_t MAT = (size_t)M_ROWS * D; // 262144 floats
    float* ws_u   = (float*)d_ws;
    float* ws_t1  = ws_u   + MAT;
    float* ws_hi  = ws_t1  + MAT;
    float* ws_hj  = ws_hi  + MAT;
    float* ws_agg = ws_hj  + MAT;          // 1024 floats
    float* ws_upd = ws_agg + 1024;         // 1024 floats

    // 1) agg = mean over nodes
    agg_kernel<<<4, 256, 0, stream>>>(nf, ws_agg);
    // 2) upd MLP on agg
    upd_mlp_kernel<<<NB, 256, 0, stream>>>(ws_agg, nu_w1, nu_b1, nu_w2, nu_b2, ws_upd);
    // 3) u = nf + upd
    build_u_kernel<<<1024, 256, 0, stream>>>(nf, ws_upd, ws_u);
    // 4) three 512^3 fp32-WMMA GEMMs
    dim3 ggrid(4, 8);
    gemm512_wmma_kernel<<<ggrid, 256, 0, stream>>>(ws_u, nt_w1, nt_b1, ws_t1, 1);
    gemm512_wmma_kernel<<<ggrid, 256, 0, stream>>>(ws_u, et_w1, (const float*)nullptr, ws_hi, 0);
    gemm512_wmma_kernel<<<ggrid, 256, 0, stream>>>(ws_u, et_w1 + (size_t)D * D, (const float*)nullptr, ws_hj, 0);
    // 5) node task head
    node_head_kernel<<<6, 256, 0, stream>>>(ws_t1, nt_w2, nt_b2, out_node);
    // 6) edge task head
    edge_kernel<<<M_ROWS, 256, 0, stream>>>(ws_hi, ws_hj, et_b1, et_w2, et_b2, out_edge);
}